// OctonionCharEmbedding_51479478010071
// MI455X (gfx1250) — compile-verified
//
#include <hip/hip_runtime.h>

#define WORD_LEN 16
#define TPB 256

typedef __attribute__((ext_vector_type(4))) float v4f;

// 7-dim cross product via octonion structure constants.
// triples (0-indexed): (0,1,2),(0,3,4),(0,6,5),(1,3,5),(1,4,6),(2,3,6),(2,5,4)
// For each triple (i,j,k):
//   o[k] += v[i]*x[j] - v[j]*x[i]
//   o[i] += v[j]*x[k] - v[k]*x[j]
//   o[j] += v[k]*x[i] - v[i]*x[k]
__device__ __forceinline__ void cross7(const float v[7], const float x[7], float o[7]) {
    const int T[7][3] = {{0,1,2},{0,3,4},{0,6,5},{1,3,5},{1,4,6},{2,3,6},{2,5,4}};
#pragma unroll
    for (int k = 0; k < 7; ++k) o[k] = 0.0f;
#pragma unroll
    for (int t = 0; t < 7; ++t) {
        const int i = T[t][0], j = T[t][1], k = T[t][2];
        o[k] = __builtin_fmaf(v[i], x[j], o[k]);
        o[k] = __builtin_fmaf(-v[j], x[i], o[k]);
        o[i] = __builtin_fmaf(v[j], x[k], o[i]);
        o[i] = __builtin_fmaf(-v[k], x[j], o[i]);
        o[j] = __builtin_fmaf(v[k], x[i], o[j]);
        o[j] = __builtin_fmaf(-v[i], x[k], o[j]);
    }
}

// LDS table layout: row r (32B, 16B-aligned) holds the 7 imaginary components
// at floats [0..6]; float [7] is padding. A gather is then ds_load_b128 (x0..x3,
// all lanes used) + ds_load_b96 at +16 (x4..x6, 16B-aligned) -- 2 DS ops, no
// dead components, no register pinning needed.
__device__ __forceinline__ void lds_row_gather(const float* __restrict__ semb,
                                               int id, float x[7]) {
    const float* __restrict__ row = semb + (id << 3);
    v4f lo = *(const v4f*)row;
    x[0] = lo.x; x[1] = lo.y; x[2] = lo.z; x[3] = lo.w;
    x[4] = row[4]; x[5] = row[5]; x[6] = row[6];
}

__global__ __launch_bounds__(TPB) void oct_word_emb_kernel(
    const int* __restrict__ ids,      // (batch, 16) int32
    const float* __restrict__ emb,    // (256, 8) f32
    float* __restrict__ out,          // (batch, 7) f32
    int batch)
{
    __shared__ __align__(16) float semb[256 * 8];

    // ---- Stage the embedding table into LDS via CDNA5 async global->LDS DMA,
    // shifted left by one float so imag components start at each 16B row base.
    // INST_OFFSET applies to BOTH the global and LDS address, so one address
    // pair + 7 offset variants copies emb[r*8+1+c] -> semb[r*8+c], c=0..6.
    {
        const unsigned t    = threadIdx.x;              // 256 threads -> one row each
        const unsigned goff = t * 32u + 4u;             // skip the real component
        const unsigned loff = (unsigned)(unsigned long long)(uintptr_t)(&semb[t * 8]);
        const unsigned long long gbase = (unsigned long long)(uintptr_t)emb;
        asm volatile(
            "global_load_async_to_lds_b32 %0, %1, %2\n\t"
            "global_load_async_to_lds_b32 %0, %1, %2 offset:4\n\t"
            "global_load_async_to_lds_b32 %0, %1, %2 offset:8\n\t"
            "global_load_async_to_lds_b32 %0, %1, %2 offset:12\n\t"
            "global_load_async_to_lds_b32 %0, %1, %2 offset:16\n\t"
            "global_load_async_to_lds_b32 %0, %1, %2 offset:20\n\t"
            "global_load_async_to_lds_b32 %0, %1, %2 offset:24\n\t"
            "s_wait_asynccnt 0"
            :
            : "v"(loff), "v"(goff), "s"(gbase)
            : "memory");
    }
    __syncthreads();

    const unsigned b = blockIdx.x * TPB + threadIdx.x;
    if ((int)b >= batch) return;

    // ---- Load this word's 16 character ids with 4 x b128 loads (64B/thread, coalesced).
    const int4* __restrict__ idv = (const int4*)(ids + (size_t)b * WORD_LEN);
    int4 w0 = idv[0];
    int4 w1 = idv[1];
    int4 w2 = idv[2];
    int4 w3 = idv[3];
    int id[WORD_LEN] = {w0.x, w0.y, w0.z, w0.w,
                        w1.x, w1.y, w1.z, w1.w,
                        w2.x, w2.y, w2.z, w2.w,
                        w3.x, w3.y, w3.z, w3.w};

    // init = imag part of first character
    float v[7];
    lds_row_gather(semb, id[0], v);

    // fold 15 cross-product steps: v = cross7(v, x_s)
#pragma unroll
    for (int s = 1; s < WORD_LEN; ++s) {
        float x[7];
        lds_row_gather(semb, id[s], x);
        float o[7];
        cross7(v, x, o);
#pragma unroll
        for (int k = 0; k < 7; ++k) v[k] = o[k];
    }

    float* __restrict__ op = out + (size_t)b * 7;
#pragma unroll
    for (int k = 0; k < 7; ++k) op[k] = v[k];
}

extern "C" void kernel_launch(void* const* d_in, const int* in_sizes, int n_in,
                              void* d_out, int out_size, void* d_ws, size_t ws_size,
                              hipStream_t stream) {
    (void)n_in; (void)out_size; (void)d_ws; (void)ws_size;
    const int*   ids = (const int*)d_in[0];
    const float* emb = (const float*)d_in[1];
    float*       out = (float*)d_out;

    const int batch  = in_sizes[0] / WORD_LEN;   // 524288
    const int blocks = (batch + TPB - 1) / TPB;  // 2048

    oct_word_emb_kernel<<<blocks, TPB, 0, stream>>>(ids, emb, out, batch);
}